// LBPModule_21010980012167
// MI455X (gfx1250) — compile-verified
//
#include <hip/hip_runtime.h>

typedef __attribute__((ext_vector_type(2))) float v2f;
typedef __attribute__((ext_vector_type(8))) float v8f;

#define NPIX  4194304          // 4*16*256*256 (B*F*H*W)
#define NFR   64               // B*F frames

// ---------------- pass 0: zero per-frame max ----------------
__global__ void k_init_max(int* __restrict__ mx) {
    if (threadIdx.x < NFR) mx[threadIdx.x] = 0;
}

// ---------------- pass 1: gray = floor(clip(mean(x)*255)) as u8 ----------------
__global__ __launch_bounds__(256) void k_gray(const float* __restrict__ x,
                                              unsigned char* __restrict__ gray) {
    int idx = blockIdx.x * blockDim.x + threadIdx.x;
    if (idx >= NPIX) return;
    int b   = idx >> 20;                     // F*H*W = 2^20
    int rem = idx & ((1 << 20) - 1);
    const float* xb = x + ((size_t)b * 3 << 20) + rem;
    float s = xb[0] + xb[1 << 20] + xb[2 << 20];
    float m = s * (1.0f / 3.0f);
    float v = floorf(fminf(fmaxf(m * 255.0f, 0.0f), 255.0f));
    gray[idx] = (unsigned char)v;
}

// ---------------- pass 2: uniform LBP + per-frame max ----------------
__global__ __launch_bounds__(256) void k_lbp(const unsigned char* __restrict__ gray,
                                             float* __restrict__ lbp,
                                             int* __restrict__ mx) {
    int idx = blockIdx.x * blockDim.x + threadIdx.x;
    if (idx >= NPIX) return;
    int frame = idx >> 16;
    int pix   = idx & 65535;
    int h = pix >> 8, w = pix & 255;
    const unsigned char* g0 = gray + ((size_t)frame << 16);

    float g[3][3];
#pragma unroll
    for (int dr = -1; dr <= 1; ++dr)
#pragma unroll
        for (int dc = -1; dc <= 1; ++dc) {
            int hh = h + dr, ww = w + dc;
            g[dr + 1][dc + 1] = (hh >= 0 && hh < 256 && ww >= 0 && ww < 256)
                                    ? (float)g0[(hh << 8) + ww] : 0.0f;
        }
    // bilinear weights for |offset| = 1/sqrt(2): t=0.70710678, a=1-t
    const float TT = 0.5f, TA = 0.20710678f, AA = 0.08578644f;
    float c  = g[1][1];
    float n0 = g[1][2];
    float n1 = TA * g[0][1] + TT * g[0][2] + AA * g[1][1] + TA * g[1][2];
    float n2 = g[0][1];
    float n3 = TT * g[0][0] + TA * g[0][1] + TA * g[1][0] + AA * g[1][1];
    float n4 = g[1][0];
    float n5 = TA * g[1][0] + AA * g[1][1] + TT * g[2][0] + TA * g[2][1];
    float n6 = g[2][1];
    float n7 = AA * g[1][1] + TA * g[1][2] + TA * g[2][1] + TT * g[2][2];

    int s0 = n0 >= c, s1 = n1 >= c, s2 = n2 >= c, s3 = n3 >= c;
    int s4 = n4 >= c, s5 = n5 >= c, s6 = n6 >= c, s7 = n7 >= c;
    int changes = abs(s0 - s1) + abs(s1 - s2) + abs(s2 - s3) + abs(s3 - s4)
                + abs(s4 - s5) + abs(s5 - s6) + abs(s6 - s7);
    int ones = s0 + s1 + s2 + s3 + s4 + s5 + s6 + s7;
    int v = (changes <= 2) ? ones : 9;
    lbp[idx] = (float)v;
    atomicMax(&mx[frame], v);
}

// ---------------- pass 3: per-frame normalize in place ----------------
__global__ __launch_bounds__(256) void k_norm(float* __restrict__ lbp,
                                              const int* __restrict__ mx) {
    int idx = blockIdx.x * blockDim.x + threadIdx.x;
    if (idx >= NPIX) return;
    int m = mx[idx >> 16];
    float sc = (m > 0) ? (1.0f / (float)m) : 1.0f;
    lbp[idx] *= sc;
}

// ---------------- pass 4: implicit-GEMM conv via V_WMMA_F32_16X16X4_F32 ----------------
// Per wave: 16x16 tile, M = 16 w-positions, N = 16 (oc 0..7 valid), K = 28 (27 taps).
// Block: 256 threads = 8 waves = 8 consecutive h rows. Grid: (W/16, H/8, B*F).
__global__ __launch_bounds__(256) void k_conv_wmma(const float* __restrict__ lbp,
                                                   const float* __restrict__ cw,
                                                   float* __restrict__ out) {
    __shared__ float WB[28][16];     // WB[k][oc] = sum_ic conv_w[oc,ic,k], zero padded
    int tid = threadIdx.x;
    for (int t = tid; t < 28 * 16; t += 256) {
        int k = t >> 4, n = t & 15;
        float v = 0.0f;
        if (k < 27 && n < 8)
            v = cw[n * 81 + k] + cw[n * 81 + 27 + k] + cw[n * 81 + 54 + k];
        WB[k][n] = v;
    }
    __syncthreads();

    int frame = blockIdx.z;
    int b = frame >> 4, f = frame & 15;
    int w0   = blockIdx.x << 4;
    int wave = tid >> 5;
    int h    = (blockIdx.y << 3) + wave;
    int l    = tid & 31;
    int m    = l & 15;                 // A row (w position in tile)
    int ks   = (l >> 4) << 1;          // K sub-offset: lanes 0-15 -> K0,1; 16-31 -> K2,3
    int n    = l & 15;                 // B/D column (output channel)

    const float* lf = lbp + ((size_t)frame << 16);
    v8f acc = {};
#pragma unroll
    for (int k4 = 0; k4 < 7; ++k4) {
        v2f a, bm;
#pragma unroll
        for (int j = 0; j < 2; ++j) {
            int kk = (k4 << 2) + ks + j;       // tap index 0..27
            float av = 0.0f;
            if (kk < 27) {
                int dd = kk / 9;
                int r9 = kk - dd * 9;
                int dh = r9 / 3;
                int dw = r9 - dh * 3;
                int ff = f + dd - 1;
                int hh = h + dh - 1;
                int ww = w0 + m + dw - 1;
                if (ff >= 0 && ff < 16 && hh >= 0 && hh < 256 && ww >= 0 && ww < 256)
                    av = lf[((dd - 1) << 16) + (hh << 8) + ww];
            }
            a[j]  = av;
            bm[j] = WB[kk & 31][n];            // kk <= 27 < 28 rows; &31 keeps idx in range
        }
        // D = A(16x4 f32) * B(4x16 f32) + C  -> v_wmma_f32_16x16x4_f32
        acc = __builtin_amdgcn_wmma_f32_16x16x4_f32(
            /*neg_a=*/false, a, /*neg_b=*/false, bm,
            /*c_mod=*/(short)0, acc, /*reuse_a=*/false, /*reuse_b=*/false);
    }

    // C/D layout: lanes 0-15 hold M=r (VGPR r), N=lane; lanes 16-31 hold M=8+r, N=lane-16
    int oc = n;
    if (oc < 8) {
        int wbase = w0 + ((l >> 4) << 3);
        float* op = out + (((size_t)(b * 8 + oc) * 16 + f) << 16) + (h << 8) + wbase;
#pragma unroll
        for (int r = 0; r < 8; ++r) op[r] = fmaxf(acc[r], 0.0f);
    }
}

extern "C" void kernel_launch(void* const* d_in, const int* in_sizes, int n_in,
                              void* d_out, int out_size, void* d_ws, size_t ws_size,
                              hipStream_t stream) {
    const float* x  = (const float*)d_in[0];   // [4,3,16,256,256]
    const float* cw = (const float*)d_in[1];   // [8,3,3,3,3]
    float* out = (float*)d_out;                // [4,8,16,256,256]

    // workspace layout: lbp f32 (16 MiB) | gray u8 (4 MiB) | per-frame max (256 B)
    float*         lbp  = (float*)d_ws;
    unsigned char* gray = (unsigned char*)d_ws + (size_t)NPIX * sizeof(float);
    int*           mxb  = (int*)(gray + NPIX);

    int blocks = NPIX / 256;
    k_init_max<<<1, 64, 0, stream>>>(mxb);
    k_gray<<<blocks, 256, 0, stream>>>(x, gray);
    k_lbp<<<blocks, 256, 0, stream>>>(gray, lbp, mxb);
    k_norm<<<blocks, 256, 0, stream>>>(lbp, mxb);
    k_conv_wmma<<<dim3(16, 32, 64), 256, 0, stream>>>(lbp, cw, out);
}